// CWFA_AO_58420145160280
// MI455X (gfx1250) — compile-verified
//
#include <hip/hip_runtime.h>

typedef __attribute__((ext_vector_type(2))) float v2f;
typedef __attribute__((ext_vector_type(4))) float v4f;
typedef __attribute__((ext_vector_type(8))) float v8f;

#define T_STEPS 128
#define RAW 16
#define KDIM 1024          // p dimension = R*DA ; q dimension = DO*R
#define SMEM_FLOATS 19552  // part(16384)+aenc(512)+oenc(512)+tmp(1024)+wa(512)+wo(512)+ba/bo/om(96)
#define SMEM_BYTES (SMEM_FLOATS * 4)
#define PACK_BYTES (KDIM * KDIM * 4)  // 4 MB packed-B image of A

// One-time repack of A[i][j][k][l] (== A2[p][q], p=i*32+j, q=k*32+l) into the
// per-lane WMMA B-operand layout: Bpk[w][kk][lane][4] =
//   { A2[p0][q], A2[p0+1][q], A2[p0][q+16], A2[p0+1][q+16] },
//   p0 = 4*kk + 2*(lane>>4), q = w*32 + (lane&15).
// One global_load_b128 then feeds both q-tiles of wave w at k-step kk.
__global__ __launch_bounds__(256)
void cwfa_pack_kernel(const float* __restrict__ A2, float* __restrict__ Bpk)
{
    const int g    = blockIdx.x * 256 + threadIdx.x;   // 0 .. 256K-1 groups
    const int lane = g & 31;
    const int kk   = (g >> 5) & 255;
    const int w    = g >> 13;                          // 0..31
    const int hi   = lane >> 4;
    const int nc   = lane & 15;
    const int p0   = 4 * kk + 2 * hi;
    const int q    = w * 32 + nc;
    v4f v;
    v.x = A2[(size_t)p0 * KDIM + q];
    v.y = A2[(size_t)(p0 + 1) * KDIM + q];
    v.z = A2[(size_t)p0 * KDIM + q + 16];
    v.w = A2[(size_t)(p0 + 1) * KDIM + q + 16];
    *(v4f*)(Bpk + (size_t)g * 4) = v;
}

template <bool PACKED>
__global__ __launch_bounds__(1024)
void cwfa_ao_kernel(const float* __restrict__ actions,
                    const float* __restrict__ obss,
                    const float* __restrict__ Wa,
                    const float* __restrict__ ba,
                    const float* __restrict__ bo,
                    const float* __restrict__ Wo,
                    const float* __restrict__ alpha,
                    const float* __restrict__ A2,    // raw A (unpacked B path)
                    const float* __restrict__ Bpk,   // packed B (PACKED path)
                    const float* __restrict__ Omega,
                    float* __restrict__ out)
{
    extern __shared__ float smem[];
    float* s_part = smem;                 // [32][16*32] per-wave k-partials
    float* s_aenc = smem + 16384;         // [16*32]
    float* s_oenc = s_aenc + 512;         // [16*32]
    float* s_tmp0 = s_oenc + 512;         // [16*32] state buffers
    float* s_tmp1 = s_tmp0 + 512;
    float* s_wa   = s_tmp1 + 512;         // [16*32]
    float* s_wo   = s_wa + 512;           // [16*32]
    float* s_ba   = s_wo + 512;           // [32]
    float* s_bo   = s_ba + 32;            // [32]
    float* s_om   = s_bo + 32;            // [32]

    const int tid = threadIdx.x;
    const int n0  = blockIdx.x * 16;

    // one-time staging of weights / initial state
    if (tid < 512) {
        s_wa[tid]   = Wa[tid];
        s_tmp0[tid] = alpha[tid & 31];    // s_0[n,i] = alpha[i]
    } else {
        s_wo[tid - 512] = Wo[tid - 512];
    }
    if (tid < 32) {
        s_ba[tid] = ba[tid];
        s_bo[tid] = bo[tid];
        s_om[tid] = Omega[tid];
    }
    __syncthreads();

    const int wave = tid >> 5;            // 0..31 == k value owned by this wave
    const int lane = tid & 31;
    const int hi   = lane >> 4;
    const int nc   = lane & 15;
    const int koff = hi * 2;
    const int row  = nc;                  // trajectory row for A operand

    for (int t = 0; t < T_STEPS; ++t) {
        // ---- encoders for step t (1024 threads, one output each) ----
        {
            const int enc = tid >> 9;            // 0: action, 1: obs
            const int nn  = (tid >> 5) & 15;
            const int jj  = tid & 31;
            const float* __restrict__ src = enc ? obss : actions;
            const float* __restrict__ W   = enc ? s_wo : s_wa;
            float s = (enc ? s_bo : s_ba)[jj];
            const size_t base = ((size_t)(n0 + nn) * T_STEPS + t) * RAW;
            #pragma unroll
            for (int r = 0; r < RAW; ++r)
                s += src[base + r] * W[r * 32 + jj];
            (enc ? s_oenc : s_aenc)[nn * 32 + jj] = s;
        }
        // ---- fixed-order reduction of previous partials -> s_t (deterministic) ----
        float* curbuf = (t & 1) ? s_tmp1 : s_tmp0;
        if (t > 0 && tid < 512) {
            float s = 0.f;
            #pragma unroll
            for (int w = 0; w < 32; ++w) s += s_part[w * 512 + tid];
            curbuf[tid] = s;
        }
        __syncthreads();

        // ---- one recurrence step on the matrix pipe ----
        v8f acc0 = {}, acc1 = {};
        #pragma unroll 4
        for (int kk = 0; kk < 256; ++kk) {
            const int p0 = kk * 4 + koff;
            // A operand: X[row, p0+c] = s_t[row, p>>5] * a_t[row, p&31]; p>>5 == kk>>3
            const float tv = curbuf[row * 32 + (kk >> 3)];
            const float2 av = *(const float2*)(s_aenc + row * 32 + (p0 & 31));
            v2f Aop; Aop.x = tv * av.x; Aop.y = tv * av.y;
            v2f B0, B1;
            if constexpr (PACKED) {
                const v4f bv = *(const v4f*)(Bpk +
                    ((((size_t)wave * 256 + kk) * 32 + lane) * 4));
                B0.x = bv.x; B0.y = bv.y;
                B1.x = bv.z; B1.y = bv.w;
            } else {
                const float* __restrict__ bp = A2 + (size_t)p0 * KDIM + wave * 32 + nc;
                B0.x = bp[0];  B0.y = bp[KDIM];
                B1.x = bp[16]; B1.y = bp[KDIM + 16];
            }
            acc0 = __builtin_amdgcn_wmma_f32_16x16x4_f32(false, Aop, false, B0, (short)0, acc0, false, false);
            acc1 = __builtin_amdgcn_wmma_f32_16x16x4_f32(false, Aop, false, B1, (short)0, acc1, false, false);
        }
        // ---- contract with o_t (k = wave) -> private partial slab ----
        #pragma unroll
        for (int r = 0; r < 8; ++r) {
            const int n = hi * 8 + r;            // D layout: VGPR r -> M=r / M=8+r
            const float o = s_oenc[n * 32 + wave];
            s_part[wave * 512 + n * 32 + nc]      = acc0[r] * o;
            s_part[wave * 512 + n * 32 + 16 + nc] = acc1[r] * o;
        }
        __syncthreads();
    }

    // ---- final state + output projection: out[n] = s_T[n,:] @ Omega ----
    if (tid < 512) {
        float s = 0.f;
        #pragma unroll
        for (int w = 0; w < 32; ++w) s += s_part[w * 512 + tid];
        s_tmp0[tid] = s;
    }
    __syncthreads();
    if (tid < 16) {
        float s = 0.f;
        #pragma unroll
        for (int l = 0; l < 32; ++l) s += s_tmp0[tid * 32 + l] * s_om[l];
        out[n0 + tid] = s;
    }
}

extern "C" void kernel_launch(void* const* d_in, const int* in_sizes, int n_in,
                              void* d_out, int out_size, void* d_ws, size_t ws_size,
                              hipStream_t stream) {
    (void)in_sizes; (void)n_in; (void)out_size;
    const float* actions = (const float*)d_in[0];
    const float* obss    = (const float*)d_in[1];
    const float* Wa      = (const float*)d_in[2];
    const float* ba      = (const float*)d_in[3];
    const float* Wo      = (const float*)d_in[4];
    const float* bo      = (const float*)d_in[5];
    const float* alpha   = (const float*)d_in[6];
    const float* A2      = (const float*)d_in[7];
    const float* Omega   = (const float*)d_in[8];
    float* out = (float*)d_out;

    const bool packed = (d_ws != nullptr) && (ws_size >= (size_t)PACK_BYTES);

    // Allow >64KB dynamic LDS (gfx1250 WGP has 320KB). Host-side, capture-safe.
    (void)hipFuncSetAttribute((const void*)&cwfa_ao_kernel<true>,
                              hipFuncAttributeMaxDynamicSharedMemorySize, SMEM_BYTES);
    (void)hipFuncSetAttribute((const void*)&cwfa_ao_kernel<false>,
                              hipFuncAttributeMaxDynamicSharedMemorySize, SMEM_BYTES);

    if (packed) {
        float* Bpk = (float*)d_ws;
        cwfa_pack_kernel<<<dim3(1024), dim3(256), 0, stream>>>(A2, Bpk);
        cwfa_ao_kernel<true><<<dim3(16), dim3(1024), SMEM_BYTES, stream>>>(
            actions, obss, Wa, ba, bo, Wo, alpha, A2, Bpk, Omega, out);
    } else {
        cwfa_ao_kernel<false><<<dim3(16), dim3(1024), SMEM_BYTES, stream>>>(
            actions, obss, Wa, ba, bo, Wo, alpha, A2, nullptr, Omega, out);
    }
}